// scHetG_71107478553104
// MI455X (gfx1250) — compile-verified
//
#include <hip/hip_runtime.h>
#include <math.h>

// scHetG pipeline for MI455X (gfx1250, wave32).
// Bandwidth-bound SPMM/gather stages in fp32 with global float atomics
// (everything hot is L2-resident: 192MB L2 >> 13MB working set).
// Dense emb layer uses V_WMMA_F32_16X16X4_F32 (fp32 matrix path).

typedef float v2f __attribute__((ext_vector_type(2)));
typedef float v8f __attribute__((ext_vector_type(8)));

namespace {
constexpr int kNC = 50000;
constexpr int kNG = 2000;
constexpr int kD  = 64;
constexpr int kNE = 1000000;
constexpr int kNP = 500000;
constexpr float kW = 1.0f / 3.0f;
constexpr float kBnEps = 1e-5f;
}

// ---------------------------------------------------------------- utilities

__global__ void zero_f32(float* __restrict__ p, size_t n) {
  size_t i = (size_t)blockIdx.x * blockDim.x + threadIdx.x;
  size_t stride = (size_t)gridDim.x * blockDim.x;
  for (; i < n; i += stride) p[i] = 0.0f;
}

__global__ void scale_copy(const float* __restrict__ in, float* __restrict__ out,
                           float w, size_t n) {
  size_t i = (size_t)blockIdx.x * blockDim.x + threadIdx.x;
  if (i < n) out[i] = w * in[i];
}

// ---------------------------------------------------------------- degrees

__global__ void deg_count(const int* __restrict__ s1, const int* __restrict__ d1,
                          const int* __restrict__ s2, const int* __restrict__ d2,
                          float* __restrict__ dc1, float* __restrict__ dg1,
                          float* __restrict__ dc2, float* __restrict__ dg2, int ne) {
  int e = blockIdx.x * blockDim.x + threadIdx.x;
  if (e >= ne) return;
  atomicAdd(&dc1[s1[e]], 1.0f);
  atomicAdd(&dg1[d1[e]], 1.0f);
  atomicAdd(&dc2[s2[e]], 1.0f);
  atomicAdd(&dg2[d2[e]], 1.0f);
}

// deg -> deg^{-1/2} (0 for isolated), two arrays at once
__global__ void inv_sqrt2(float* __restrict__ a, float* __restrict__ b, int n) {
  int i = blockIdx.x * blockDim.x + threadIdx.x;
  if (i >= n) return;
  float x = a[i]; a[i] = (x > 0.0f) ? rsqrtf(x) : 0.0f;
  float y = b[i]; b[i] = (y > 0.0f) ? rsqrtf(y) : 0.0f;
}

// ---------------------------------------------------------------- SPMM

// One thread per (edge, feature). Does BOTH directions of one relation:
//   gene accumulator += cj[s] * cell[s][f]   (cell -> gene)
//   cell accumulator += ci[d] * gene[d][f]   (gene -> cell)
__global__ void spmm_both(const float* __restrict__ cfeat, const float* __restrict__ gfeat,
                          const float* __restrict__ cj, const float* __restrict__ ci,
                          const int* __restrict__ src, const int* __restrict__ dst,
                          float* __restrict__ gacc, float* __restrict__ cacc, int ne) {
  size_t gid = (size_t)blockIdx.x * blockDim.x + threadIdx.x;
  size_t e = gid >> 6;
  int f = (int)(gid & 63);
  if (e >= (size_t)ne) return;
  int s = src[e];
  int d = dst[e];
  atomicAdd(&gacc[(size_t)d * kD + f], cfeat[(size_t)s * kD + f] * cj[s]);
  atomicAdd(&cacc[(size_t)s * kD + f], gfeat[(size_t)d * kD + f] * ci[d]);
}

// gnext = 0.5*(ci1*acc1 + ci2*acc2); ih += w*gnext
__global__ void finish_gene(const float* __restrict__ ga1, const float* __restrict__ ga2,
                            const float* __restrict__ ci1, const float* __restrict__ ci2,
                            float* __restrict__ gnext, float* __restrict__ ih,
                            float w, int n) {
  int i = blockIdx.x * blockDim.x + threadIdx.x;
  if (i >= n) return;
  int g = i >> 6;
  float v = 0.5f * (ci1[g] * ga1[i] + ci2[g] * ga2[i]);
  gnext[i] = v;
  ih[i] += w * v;
}

// cacc *= cj (in place -> becomes c_next); u += w*c_next
__global__ void finish_cell(float* __restrict__ cacc, const float* __restrict__ cj,
                            float* __restrict__ u, float w, size_t n) {
  size_t i = (size_t)blockIdx.x * blockDim.x + threadIdx.x;
  if (i >= n) return;
  size_t c = i >> 6;
  float v = cj[c] * cacc[i];
  cacc[i] = v;
  u[i] += w * v;
}

// ---------------------------------------------------------------- emb layer (WMMA)

// UF = ELU( BN( [U | onehot] @ W + b ) ), W is (66 x 64) row-major.
// One wave per 16-row tile; four 16x16 f32 accumulators cover all 64 cols.
// V_WMMA_F32_16X16X4_F32 layouts (ISA 7.12.2):
//   A (16x4):  lane m=l&15;  VGPR0 = K(0|2), VGPR1 = K(1|3) split by lane half
//   B (4x16):  col n=l&15;   VGPR0 = row(0|2), VGPR1 = row(1|3) split by lane half
//   C/D:       VGPR v: lanes0-15 -> M=v, lanes16-31 -> M=v+8; N = l&15
__global__ void emb_wmma(const float* __restrict__ U, float* __restrict__ UF,
                         const float* __restrict__ W, const float* __restrict__ bias,
                         const float* __restrict__ gam, const float* __restrict__ bet,
                         const float* __restrict__ mu, const float* __restrict__ var,
                         int oh_row, int nrows) {
  unsigned gtid = blockIdx.x * blockDim.x + threadIdx.x;
  unsigned wave = gtid >> 5;          // uniform within a wave
  unsigned lane = threadIdx.x & 31;
  int row0 = (int)wave * 16;
  if (row0 >= nrows) return;          // wave-uniform: EXEC stays all-ones
  int m   = (int)(lane & 15);
  int hi  = (int)(lane >> 4);         // 0 or 1 (lane half)
  int sel = hi * 2;                   // K sub-select within each group of 4
  const float* urow = U + (size_t)(row0 + m) * kD;

  v8f acc0 = {}, acc1 = {}, acc2 = {}, acc3 = {};
  for (int kk = 0; kk < 16; ++kk) {   // K = 64 in steps of 4
    int kb = kk * 4 + sel;
    v2f a;
    a.x = urow[kb];
    a.y = urow[kb + 1];
    const float* w0 = W + (size_t)kb * kD + m;
    const float* w1 = W + (size_t)(kb + 1) * kD + m;
    v2f b0; b0.x = w0[0];  b0.y = w1[0];
    v2f b1; b1.x = w0[16]; b1.y = w1[16];
    v2f b2; b2.x = w0[32]; b2.y = w1[32];
    v2f b3; b3.x = w0[48]; b3.y = w1[48];
    acc0 = __builtin_amdgcn_wmma_f32_16x16x4_f32(false, a, false, b0, (short)0, acc0, false, false);
    acc1 = __builtin_amdgcn_wmma_f32_16x16x4_f32(false, a, false, b1, (short)0, acc1, false, false);
    acc2 = __builtin_amdgcn_wmma_f32_16x16x4_f32(false, a, false, b2, (short)0, acc2, false, false);
    acc3 = __builtin_amdgcn_wmma_f32_16x16x4_f32(false, a, false, b3, (short)0, acc3, false, false);
  }

  v8f accs[4] = {acc0, acc1, acc2, acc3};
  // all reads of U done; safe to write UF == U in place
  for (int nt = 0; nt < 4; ++nt) {
    int col = nt * 16 + m;
    float sc  = gam[col] * rsqrtf(var[col] + kBnEps);
    float sh  = bet[col] - mu[col] * sc;
    float add = bias[col] + W[(size_t)oh_row * kD + col];  // bias + onehot row of W
    v8f c = accs[nt];
#pragma unroll
    for (int v = 0; v < 8; ++v) {
      float y = (c[v] + add) * sc + sh;
      y = (y > 0.0f) ? y : (expf(y) - 1.0f);               // ELU
      UF[(size_t)(row0 + v + 8 * hi) * kD + col] = y;
    }
  }
}

// ---------------------------------------------------------------- decoder

// One wave32 per positive edge; 2 floats per lane; butterfly reduction.
__global__ void dot_decode(const float* __restrict__ uf1, const float* __restrict__ uf2,
                           const float* __restrict__ ih,
                           const int* __restrict__ ps1, const int* __restrict__ pd1,
                           const int* __restrict__ ps2, const int* __restrict__ pd2,
                           float* __restrict__ out, int np) {
  size_t gtid = (size_t)blockIdx.x * blockDim.x + threadIdx.x;
  int lane = (int)(threadIdx.x & 31);
  size_t e = gtid >> 5;
  if (e >= (size_t)2 * np) return;
  const float* uf;
  int s, d;
  if (e < (size_t)np) {
    s = ps1[e]; d = pd1[e]; uf = uf1;
  } else {
    size_t k = e - (size_t)np;
    s = ps2[k]; d = pd2[k]; uf = uf2;
  }
  const float* a = uf + (size_t)s * kD;
  const float* g = ih + (size_t)d * kD;
  float acc = a[lane] * g[lane] + a[lane + 32] * g[lane + 32];
#pragma unroll
  for (int off = 16; off > 0; off >>= 1) acc += __shfl_xor(acc, off, 32);
  if (lane == 0) out[e] = acc;
}

// ---------------------------------------------------------------- launch

extern "C" void kernel_launch(void* const* d_in, const int* in_sizes, int n_in,
                              void* d_out, int out_size, void* d_ws, size_t ws_size,
                              hipStream_t stream) {
  const float* cf1 = (const float*)d_in[0];
  const float* cf2 = (const float*)d_in[1];
  const float* gf  = (const float*)d_in[2];
  const float* W   = (const float*)d_in[3];
  const float* bb  = (const float*)d_in[4];
  const float* gam = (const float*)d_in[5];
  const float* bet = (const float*)d_in[6];
  const float* mu  = (const float*)d_in[7];
  const float* var = (const float*)d_in[8];
  const int* es1 = (const int*)d_in[9];
  const int* ed1 = (const int*)d_in[10];
  const int* es2 = (const int*)d_in[11];
  const int* ed2 = (const int*)d_in[12];
  const int* ps1 = (const int*)d_in[13];
  const int* pd1 = (const int*)d_in[14];
  const int* ps2 = (const int*)d_in[15];
  const int* pd2 = (const int*)d_in[16];
  float* out = (float*)d_out;

  // workspace layout (floats), ~80 MB total
  const size_t NCD = (size_t)kNC * kD;   // 3.2M
  const size_t NGD = (size_t)kNG * kD;   // 128K
  float* ws  = (float*)d_ws;
  float* cj1 = ws;                  // NC   deg^-1/2 of enc_src1 (cells)
  float* cj2 = cj1 + kNC;           // NC
  float* ci1 = cj2 + kNC;           // NG   deg^-1/2 of enc_dst1 (genes)
  float* ci2 = ci1 + kNG;           // NG
  float* c1A = ci2 + kNG;           // NCD  cell state ping/pong + accumulators
  float* c2A = c1A + NCD;
  float* c1B = c2A + NCD;
  float* c2B = c1B + NCD;
  float* u1  = c2B + NCD;           // NCD  running sum -> uf1 (in place)
  float* u2  = u1 + NCD;            // NCD  running sum -> uf2 (in place)
  float* gA  = u2 + NCD;            // NGD  gene state ping/pong
  float* gB  = gA + NGD;
  float* ga1 = gB + NGD;            // NGD  gene accumulators (per relation)
  float* ga2 = ga1 + NGD;
  float* ih  = ga2 + NGD;           // NGD  running gene sum

  auto nb = [](size_t n, int b) { return (unsigned)((n + (size_t)b - 1) / (size_t)b); };

  // 1. degree normalizers
  zero_f32<<<256, 256, 0, stream>>>(cj1, (size_t)(2 * kNC + 2 * kNG));
  deg_count<<<nb(kNE, 256), 256, 0, stream>>>(es1, ed1, es2, ed2, cj1, ci1, cj2, ci2, kNE);
  inv_sqrt2<<<nb(kNC, 256), 256, 0, stream>>>(cj1, cj2, kNC);
  inv_sqrt2<<<nb(kNG, 256), 256, 0, stream>>>(ci1, ci2, kNG);

  // 2. init running sums: u = w*feat
  scale_copy<<<nb(NCD, 256), 256, 0, stream>>>(cf1, u1, kW, NCD);
  scale_copy<<<nb(NCD, 256), 256, 0, stream>>>(cf2, u2, kW, NCD);
  scale_copy<<<nb(NGD, 256), 256, 0, stream>>>(gf, ih, kW, NGD);

  // 3. two GNN layers
  const float* curc1 = cf1;
  const float* curc2 = cf2;
  const float* curg  = gf;
  float* nc1 = c1A; float* nc2 = c2A; float* ng = gA;
  const size_t spmm_threads = (size_t)kNE * kD;  // 64M
  for (int it = 0; it < 2; ++it) {
    zero_f32<<<256, 256, 0, stream>>>(ga1, NGD);
    zero_f32<<<256, 256, 0, stream>>>(ga2, NGD);
    zero_f32<<<1024, 256, 0, stream>>>(nc1, NCD);
    zero_f32<<<1024, 256, 0, stream>>>(nc2, NCD);
    spmm_both<<<nb(spmm_threads, 256), 256, 0, stream>>>(curc1, curg, cj1, ci1,
                                                         es1, ed1, ga1, nc1, kNE);
    spmm_both<<<nb(spmm_threads, 256), 256, 0, stream>>>(curc2, curg, cj2, ci2,
                                                         es2, ed2, ga2, nc2, kNE);
    finish_gene<<<nb(NGD, 256), 256, 0, stream>>>(ga1, ga2, ci1, ci2, ng, ih, kW, (int)NGD);
    finish_cell<<<nb(NCD, 256), 256, 0, stream>>>(nc1, cj1, u1, kW, NCD);
    finish_cell<<<nb(NCD, 256), 256, 0, stream>>>(nc2, cj2, u2, kW, NCD);
    curc1 = nc1; curc2 = nc2; curg = ng;
    nc1 = c1B; nc2 = c2B; ng = gB;
  }

  // 4. emb layer (WMMA fp32), in place u -> uf
  {
    unsigned tiles = kNC / 16;              // 3125, exact
    unsigned nthreads = tiles * 32;         // one wave per tile
    emb_wmma<<<nb(nthreads, 256), 256, 0, stream>>>(u1, u1, W, bb, gam, bet, mu, var, 64, kNC);
    emb_wmma<<<nb(nthreads, 256), 256, 0, stream>>>(u2, u2, W, bb, gam, bet, mu, var, 65, kNC);
  }

  // 5. dot decoder: one wave per edge, concat(pred1, pred2)
  {
    size_t nthreads = (size_t)2 * kNP * 32;
    dot_decode<<<nb(nthreads, 256), 256, 0, stream>>>(u1, u2, ih, ps1, pd1, ps2, pd2, out, kNP);
  }
}